// STFTLayer_103079215679
// MI455X (gfx1250) — compile-verified
//
#include <hip/hip_runtime.h>
#include <math.h>

#define B_ 256
#define S_ 128
#define H_ 256
#define NF 65            // S/2+1
#define LN_EPS 1e-12f
#define PI_F 3.14159265358979323846f

typedef __attribute__((ext_vector_type(2))) float v2f;
typedef __attribute__((ext_vector_type(8))) float v8f;

// ---- workspace layout (in floats) ----
// [0, H*S*B)              : seq_fft raw, layout (H, S, B)
// [WS_G, +H*S)            : g_h[d] circulant kernel for seq_fft
// [WS_K1..WS_K4)          : branch taps k_h (n = 3,5,7,9)
#define WS_FFT 0
#define WS_G   (H_*S_*B_)
#define WS_K1  (WS_G + H_*S_)
#define WS_K2  (WS_K1 + H_*3)
#define WS_K3  (WS_K2 + H_*5)
#define WS_K4  (WS_K3 + H_*7)
// total floats = WS_K4 + H_*9  (~8.43M floats ~= 33.7 MB)

// ------------------------------------------------------------------
// Pass 0: build real convolution kernels from the complex weights.
//   g_h[d]  = (1/S)[cw0_re + 2*sum_{f=1..63}(re*cos - im*sin) + cw64_re*(-1)^d]
//   k_h[j]  = (1/n)[w0_re  + 2*sum_{f=1..F-1}(re*cos - im*sin)]   (n odd)
// ------------------------------------------------------------------
__global__ __launch_bounds__(128) void precompute_kernel(
    const float* __restrict__ cw,   // (NF, H, 2)
    const float* __restrict__ w3,   // (H, 2, 2)
    const float* __restrict__ w5,   // (H, 3, 2)
    const float* __restrict__ w7,   // (H, 4, 2)
    const float* __restrict__ w9,   // (H, 5, 2)
    float* __restrict__ ws)
{
    const int h = blockIdx.x;
    const int d = threadIdx.x;          // 0..127

    // ---- g_h[d] for the global FFT branch ----
    {
        float acc = cw[(0 * H_ + h) * 2 + 0];
        const float step = 2.f * PI_F * (float)d / (float)S_;
        for (int f = 1; f < NF - 1; ++f) {
            const float th  = step * (float)f;
            const float cre = cw[(f * H_ + h) * 2 + 0];
            const float cim = cw[(f * H_ + h) * 2 + 1];
            acc += 2.f * (cre * __cosf(th) - cim * __sinf(th));
        }
        acc += cw[((NF - 1) * H_ + h) * 2 + 0] * ((d & 1) ? -1.f : 1.f);
        ws[WS_G + h * S_ + d] = acc * (1.f / (float)S_);
    }

    // ---- k_h[j] taps for the four STFT branches (24 taps per h) ----
    if (d < 24) {
        int i, j, base;
        if      (d < 3)  { i = 1; j = d;      base = WS_K1; }
        else if (d < 8)  { i = 2; j = d - 3;  base = WS_K2; }
        else if (d < 15) { i = 3; j = d - 8;  base = WS_K3; }
        else             { i = 4; j = d - 15; base = WS_K4; }
        const float* w = (i == 1) ? w3 : (i == 2) ? w5 : (i == 3) ? w7 : w9;
        const int n = 2 * i + 1, F = i + 1;
        float acc = w[(h * F + 0) * 2 + 0];
        for (int f = 1; f < F; ++f) {
            const float th = 2.f * PI_F * (float)(f * j) / (float)n;
            acc += 2.f * (w[(h * F + f) * 2 + 0] * __cosf(th)
                        - w[(h * F + f) * 2 + 1] * __sinf(th));
        }
        ws[base + h * n + j] = acc / (float)n;
    }
}

// ------------------------------------------------------------------
// Pass 1: per-channel circulant matmul  R_h = G_h (SxS) * X_h (SxB)
// via V_WMMA_F32_16X16X4_F32.  One wave per 16x64 output strip:
// 4 accumulators share each A fragment (circulant rows of g), so each
// K-step issues 4 WMMAs off one A load pair.  128 WMMAs per wave.
//   A frag (16x4 f32): lane<16 -> M=lane,    K = kb+0, kb+1
//                      lane>=16 -> M=lane-16, K = kb+2, kb+3
//   B frag (4x16 f32): same K split, N = lane&15
//   C/D (16x16 f32):   VGPR v -> row t0 + v + 8*(lane>=16), col = lane&15
// ------------------------------------------------------------------
__global__ __launch_bounds__(32) void fftconv_wmma_kernel(
    const float* __restrict__ x,     // (B, S, H)
    const float* __restrict__ g,     // (H, S)
    float* __restrict__ r)           // (H, S, B)
{
    const int lane = threadIdx.x;
    const int h    = blockIdx.y;
    const int mt   = blockIdx.x & 7;      // 8 tiles over S (rows)
    const int nb   = blockIdx.x >> 3;     // 4 strips of 64 over B (cols)
    const int t0   = mt * 16;
    const int b0   = nb * 64;

    const float* gh  = g + h * S_;
    const int half   = lane >> 4;         // 0: K=kb+0/1   1: K=kb+2/3
    const int lid    = lane & 15;

    v8f c0 = {}, c1 = {}, c2 = {}, c3 = {};

    #pragma unroll 2
    for (int kb = 0; kb < S_; kb += 4) {
        const int k0 = kb + half * 2;

        // circulant A: G[t][k] = g_h[(t-k) mod S]  (shared by all 4 tiles)
        v2f a;
        a.x = gh[(t0 + lid - k0)     & (S_ - 1)];
        a.y = gh[(t0 + lid - k0 - 1) & (S_ - 1)];

        // B frags: X[k][n] = x[b0+n, k, h], four 16-wide column groups
        const float* xp = x + ((size_t)(b0 + lid) * S_ + k0) * H_ + h;
        // prefetch next K-chunk (stride-1KB gather: L0 can't help, hide L2)
        __builtin_prefetch(xp + 4 * H_, 0, 1);
        v2f bf0, bf1, bf2, bf3;
        bf0.x = xp[0];                bf0.y = xp[H_];
        bf1.x = xp[16 * S_ * H_];     bf1.y = xp[16 * S_ * H_ + H_];
        bf2.x = xp[32 * S_ * H_];     bf2.y = xp[32 * S_ * H_ + H_];
        bf3.x = xp[48 * S_ * H_];     bf3.y = xp[48 * S_ * H_ + H_];

        c0 = __builtin_amdgcn_wmma_f32_16x16x4_f32(false, a, false, bf0,
                                                   (short)0, c0, false, false);
        c1 = __builtin_amdgcn_wmma_f32_16x16x4_f32(false, a, false, bf1,
                                                   (short)0, c1, false, false);
        c2 = __builtin_amdgcn_wmma_f32_16x16x4_f32(false, a, false, bf2,
                                                   (short)0, c2, false, false);
        c3 = __builtin_amdgcn_wmma_f32_16x16x4_f32(false, a, false, bf3,
                                                   (short)0, c3, false, false);
    }

    #pragma unroll
    for (int v = 0; v < 8; ++v) {
        const int    trow = t0 + v + half * 8;
        float* rp = r + ((size_t)h * S_ + trow) * B_ + b0 + lid;
        rp[0]  = c0[v];          // coalesced 64B per VGPR row per tile
        rp[16] = c1[v];
        rp[32] = c2[v];
        rp[48] = c3[v];
    }
}

// ------------------------------------------------------------------
// Pass 2: banded STFT branches + 6 layernorms + mixing.
// Block = one (b,t) row, thread = channel h.
// Reductions: wave32 shfl butterfly + 8 LDS partials (2 barriers each).
// ------------------------------------------------------------------
__device__ __forceinline__ float block_sum(float v, float* red) {
    #pragma unroll
    for (int o = 16; o > 0; o >>= 1) v += __shfl_xor(v, o, 32);
    const int wid = threadIdx.x >> 5;           // 8 waves of 32
    if ((threadIdx.x & 31) == 0) red[wid] = v;
    __syncthreads();
    float s = 0.f;
    #pragma unroll
    for (int w = 0; w < 8; ++w) s += red[w];
    __syncthreads();
    return s;
}

__device__ __forceinline__ float lnorm(float v, float lw, float lb, float* red) {
    const float mu  = block_sum(v, red) * (1.f / (float)H_);
    const float dv  = v - mu;
    const float var = block_sum(dv * dv, red) * (1.f / (float)H_);
    return lw * dv * rsqrtf(var + LN_EPS) + lb;
}

// Banded overlap-add STFT branch, fully compile-time unrolled.
//   raw[t] = (1/env) * sum_{s=t-pad..t+pad, s in [0,S)}
//            sum_m xa[s+m-pad] * k[(t+pad-s-m) mod n]
template <int I>
__device__ __forceinline__ float branch_raw(const float* __restrict__ kp,
                                            const float* __restrict__ xa, int t) {
    constexpr int n = 2 * I + 1, pad = I;
    float kk[n];
    #pragma unroll
    for (int j = 0; j < n; ++j) kk[j] = kp[j];

    float acc = 0.f;
    int   env = 0;
    #pragma unroll
    for (int ds = -pad; ds <= pad; ++ds) {
        const int  s     = t + ds;
        const bool valid = ((unsigned)s < (unsigned)S_);
        env += valid ? 1 : 0;
        float f = 0.f;
        #pragma unroll
        for (int m = 0; m < n; ++m) {
            constexpr int nn = n;
            const int kidx = (((pad - ds - m) % nn) + nn) % nn;  // compile-time
            f += xa[ds + m - pad + 8] * kk[kidx];
        }
        acc += valid ? f : 0.f;
    }
    return acc / (float)env;
}

__global__ __launch_bounds__(256) void fuse_kernel(
    const float* __restrict__ x,       // (B, S, H)
    const float* __restrict__ ws,
    const float* __restrict__ alpha_p,
    const float* __restrict__ lnw,
    const float* __restrict__ lnb,
    float* __restrict__ out)           // (B, S, H)
{
    __shared__ float red[8];
    const int t = blockIdx.x;      // S
    const int b = blockIdx.y;      // B
    const int h = threadIdx.x;     // H

    // Load +/-8 window of x (zero-padded) — covers all branch bands (2*pad<=8).
    float xa[17];
    #pragma unroll
    for (int d = 0; d < 17; ++d) {
        const int tau = t + d - 8;
        xa[d] = ((unsigned)tau < (unsigned)S_)
                    ? x[((size_t)b * S_ + tau) * H_ + h] : 0.f;
    }

    const float lw    = lnw[h];
    const float lb    = lnb[h];
    const float alpha = alpha_p[0];

    // Four STFT branches, each layernormed then summed.
    float seq_emb = 0.f;
    seq_emb += lnorm(branch_raw<1>(ws + WS_K1 + h * 3, xa, t), lw, lb, red);
    seq_emb += lnorm(branch_raw<2>(ws + WS_K2 + h * 5, xa, t), lw, lb, red);
    seq_emb += lnorm(branch_raw<3>(ws + WS_K3 + h * 7, xa, t), lw, lb, red);
    seq_emb += lnorm(branch_raw<4>(ws + WS_K4 + h * 9, xa, t), lw, lb, red);

    // Global FFT branch (precomputed by the WMMA pass), then layernorm.
    const float fraw = ws[WS_FFT + ((size_t)h * S_ + t) * B_ + b];
    const float fln  = lnorm(fraw, lw, lb, red);

    // Mix, residual, final layernorm.
    const float mixed = alpha * seq_emb + (1.f - alpha) * fln + xa[8];
    out[((size_t)b * S_ + t) * H_ + h] = lnorm(mixed, lw, lb, red);
}

// ------------------------------------------------------------------
extern "C" void kernel_launch(void* const* d_in, const int* in_sizes, int n_in,
                              void* d_out, int out_size, void* d_ws, size_t ws_size,
                              hipStream_t stream) {
    const float* x     = (const float*)d_in[0];
    const float* w3    = (const float*)d_in[1];
    const float* w5    = (const float*)d_in[2];
    const float* w7    = (const float*)d_in[3];
    const float* w9    = (const float*)d_in[4];
    const float* cw    = (const float*)d_in[5];
    const float* alpha = (const float*)d_in[6];
    const float* lnw   = (const float*)d_in[7];
    const float* lnb   = (const float*)d_in[8];
    float* ws  = (float*)d_ws;
    float* out = (float*)d_out;

    (void)in_sizes; (void)n_in; (void)out_size; (void)ws_size;

    // Pass 0: tiny trig kernel building g and the 24 branch taps per channel.
    precompute_kernel<<<H_, 128, 0, stream>>>(cw, w3, w5, w7, w9, ws);

    // Pass 1: H per-channel 128x128x256 f32 matmuls on the WMMA path.
    //         8 S-tiles x 4 B-strips per channel, one wave each.
    fftconv_wmma_kernel<<<dim3(8 * 4, H_), 32, 0, stream>>>(
        x, ws + WS_G, ws + WS_FFT);

    // Pass 2: branches + layernorms + mix.
    fuse_kernel<<<dim3(S_, B_), 256, 0, stream>>>(x, ws, alpha, lnw, lnb, out);
}